// PointNetConv_80461917323416
// MI455X (gfx1250) — compile-verified
//
#include <hip/hip_runtime.h>

// PointNetConv fused pipeline for gfx1250 (MI455X):
//   K1 init:   accum = -inf, uv = {W1[:,64], W1[:,65]}
//   K2 wmma:   xW = x @ W1[:, :64]^T + b1          (v_wmma_f32_16x16x4_f32)
//   K3 edge:   accum[tgt] = max(accum[tgt], xW[src] + dx*u + dy*v)  (L2 atomics)
//   K4 wmma:   out = accum @ W2^T                  (v_wmma_f32_16x16x4_f32)

#define N_NODES 100000
#define N_EDGES 1600000
#define CH      64
#define LDW1    66   // W1 row stride (OUT x (IN+2))

typedef __attribute__((ext_vector_type(2))) float v2f;
typedef __attribute__((ext_vector_type(8))) float v8f;

// ---------------------------------------------------------------- init
__global__ __launch_bounds__(256)
void pnc_init_kernel(float* __restrict__ accum, float* __restrict__ uv,
                     const float* __restrict__ W1, int total) {
  int i = blockIdx.x * blockDim.x + threadIdx.x;
  if (i < total) accum[i] = __int_as_float(0xFF800000); // -inf
  if (i < 2 * CH) {
    int c = i & (CH - 1);
    int which = i >> 6;                 // 0 -> dx column, 1 -> dy column
    uv[i] = W1[c * LDW1 + CH + which];
  }
}

// ---------------------------------------------------------------- fp32 WMMA GEMM
// Y[m][n] = sum_k X[m][k] * W[n][k] (+ bias[n]).  One wave per 16-row tile,
// 4 column tiles x 16 K-steps of V_WMMA_F32_16X16X4_F32.
// A 16x4 layout: lane = M%16 + 16*h; vgpr0/1 = K = 2h, 2h+1 (per ISA 7.12.2).
// B 4x16 layout mirrors A with N striped across lanes.
__global__ __launch_bounds__(256)
void pnc_wmma_gemm_xwt(const float* __restrict__ X,
                       const float* __restrict__ W, int ldw,
                       const float* __restrict__ bias,
                       float* __restrict__ Y, int ntiles) {
  const int lane = threadIdx.x & 31;
  const int tile = blockIdx.x * (blockDim.x >> 5) + (threadIdx.x >> 5);
  if (tile >= ntiles) return;           // wave-uniform: EXEC stays all-ones
  const int h   = lane >> 4;            // K-half select
  const int l16 = lane & 15;
  const int m0  = tile * 16;

  // Preload all 16 A fragments for this 16x64 row slab.
  v2f a[16];
  const float* xrow = X + (size_t)(m0 + l16) * CH + 2 * h;
#pragma unroll
  for (int k = 0; k < 16; ++k) {
    a[k].x = xrow[4 * k + 0];
    a[k].y = xrow[4 * k + 1];
  }

#pragma unroll
  for (int nt = 0; nt < 4; ++nt) {
    const int n0 = nt * 16;
    const float* wrow = W + (size_t)(n0 + l16) * ldw + 2 * h;
    v8f c = {};
#pragma unroll
    for (int k = 0; k < 16; ++k) {
      v2f b;
      b.x = wrow[4 * k + 0];
      b.y = wrow[4 * k + 1];
      // 8 args: (neg_a, A, neg_b, B, c_mod, C, reuse_a, reuse_b)
      c = __builtin_amdgcn_wmma_f32_16x16x4_f32(
          false, a[k], false, b, (short)0, c, false, false);
    }
    const float bv = bias ? bias[n0 + l16] : 0.0f;
#pragma unroll
    for (int r = 0; r < 8; ++r) {       // C/D: vgpr r -> row r + 8*h, col l16
      Y[(size_t)(m0 + r + 8 * h) * CH + n0 + l16] = c[r] + bv;
    }
  }
}

// ---------------------------------------------------------------- edge max
// Order-independent float max via native integer atomics (L2-side, no CAS):
// positive floats compare as signed ints, negative floats reverse-compare as
// unsigned ints; -inf (0xFF800000) init is the identity for both paths.
__device__ __forceinline__ void atomic_max_f32(float* addr, float val) {
  if (val >= 0.0f) atomicMax((int*)addr, __float_as_int(val));
  else             atomicMin((unsigned int*)addr, (unsigned int)__float_as_int(val));
}

__global__ __launch_bounds__(256)
void pnc_edge_max_kernel(const int*   __restrict__ ei,
                         const float* __restrict__ pos,
                         const float* __restrict__ xW,
                         const float* __restrict__ uv,
                         float* __restrict__ accum) {
  const int t    = blockIdx.x * blockDim.x + threadIdx.x;
  const int e    = t >> 4;              // 16 lanes per edge
  const int lane = t & 15;
  if (e >= N_EDGES) return;

  const int tgt = ei[2 * e + 0];
  const int src = ei[2 * e + 1];
  const float dx = pos[3 * src + 0] - pos[3 * tgt + 0];
  const float dy = pos[3 * src + 1] - pos[3 * tgt + 1];

  const int c = lane * 4;
  const float4 xw = *(const float4*)(xW + (size_t)src * CH + c);
  const float4 u  = *(const float4*)(uv + c);
  const float4 v  = *(const float4*)(uv + CH + c);

  float4 m;
  m.x = fmaf(dy, v.x, fmaf(dx, u.x, xw.x));
  m.y = fmaf(dy, v.y, fmaf(dx, u.y, xw.y));
  m.z = fmaf(dy, v.z, fmaf(dx, u.z, xw.z));
  m.w = fmaf(dy, v.w, fmaf(dx, u.w, xw.w));

  float* dst = accum + (size_t)tgt * CH + c;
  atomic_max_f32(dst + 0, m.x);
  atomic_max_f32(dst + 1, m.y);
  atomic_max_f32(dst + 2, m.z);
  atomic_max_f32(dst + 3, m.w);
}

// ---------------------------------------------------------------- launch
extern "C" void kernel_launch(void* const* d_in, const int* in_sizes, int n_in,
                              void* d_out, int out_size, void* d_ws, size_t ws_size,
                              hipStream_t stream) {
  const float* x   = (const float*)d_in[0];   // N x 64
  const float* pos = (const float*)d_in[1];   // N x 3
  const int*   ei  = (const int*)  d_in[2];   // E x 2 (tgt, src)
  const float* W1  = (const float*)d_in[3];   // 64 x 66
  const float* b1  = (const float*)d_in[4];   // 64
  const float* W2  = (const float*)d_in[5];   // 64 x 64
  float* out = (float*)d_out;                 // N x 64

  // Workspace: [xW: N*64][accum: N*64][uv: 128]  (~51.2 MB)
  float* xW    = (float*)d_ws;
  float* accum = xW + (size_t)N_NODES * CH;
  float* uv    = accum + (size_t)N_NODES * CH;
  (void)ws_size; (void)n_in; (void)in_sizes; (void)out_size;

  const int ntiles = N_NODES / 16;            // 6250 (N divisible by 16)

  // K1: accum = -inf, uv columns
  {
    int total = N_NODES * CH;
    int blocks = (total + 255) / 256;
    pnc_init_kernel<<<blocks, 256, 0, stream>>>(accum, uv, W1, total);
  }
  // K2: xW = x @ W1[:, :64]^T + b1
  {
    int blocks = (ntiles + 7) / 8;            // 8 waves (16-row tiles) per block
    pnc_wmma_gemm_xwt<<<blocks, 256, 0, stream>>>(x, W1, LDW1, b1, xW, ntiles);
  }
  // K3: fused rank-2 edge update + segment max
  {
    int blocks = (N_EDGES * 16 + 255) / 256;  // 100000 blocks
    pnc_edge_max_kernel<<<blocks, 256, 0, stream>>>(ei, pos, xW, uv, accum);
  }
  // K4: out = accum @ W2^T
  {
    int blocks = (ntiles + 7) / 8;
    pnc_wmma_gemm_xwt<<<blocks, 256, 0, stream>>>(accum, W2, CH, nullptr, out, ntiles);
  }
}